// Predictor_89404039234103
// MI455X (gfx1250) — compile-verified
//
#include <hip/hip_runtime.h>
#include <math.h>

// ---------------- model dims ----------------
#define S_DIM 50
#define T_DIM 50
#define B_DIM 64
#define H_DIM 512
#define ES_DIM 512
#define ET_DIM 512
#define M_DIM 512
#define EO_DIM 512
#define VT_DIM 32000

typedef __attribute__((ext_vector_type(2))) float v2f;
typedef __attribute__((ext_vector_type(8))) float v8f;

// =====================================================================
// Direct fp32 WMMA GEMM:  C[M,N] (+)= A[M,K] * W[N,K]^T  (+ bias[N])
// one wave per 16x32 tile (two 16x16 WMMA accumulators share A fragment)
// Used for the small latency-bound per-step GEMMs (M == 64).
// A fragment (16x4 f32): lane L holds A[m0 + L%16][k + 2*(L/16) + {0,1}]
// B fragment (4x16 f32): lane L holds W[n0 + L%16][k + 2*(L/16) + {0,1}]
// D (16x16 f32): VGPR v, lane L -> row m0 + v + 8*(L/16), col n0 + L%16
// =====================================================================
__global__ __launch_bounds__(32)
void gemm_wmma_f32(const float* __restrict__ A, int lda,
                   const float* __restrict__ W, int ldw,
                   float* __restrict__ C, int ldc,
                   int K, const float* __restrict__ bias, int accumulate)
{
    const int lane = threadIdx.x;
    const int r    = lane & 15;
    const int kk   = (lane >> 4) << 1;     // 0 or 2
    const int m0   = blockIdx.y << 4;
    const int n0   = blockIdx.x << 5;      // 32 columns per wave

    const float* ap  = A + (size_t)(m0 + r)      * lda + kk;
    const float* wp0 = W + (size_t)(n0 + r)      * ldw + kk;
    const float* wp1 = W + (size_t)(n0 + 16 + r) * ldw + kk;

    v8f acc0 = {}; v8f acc1 = {};
#pragma unroll 4
    for (int k = 0; k < K; k += 4) {
        v2f a, b0, b1;
        a.x  = ap[k];   a.y  = ap[k + 1];
        b0.x = wp0[k];  b0.y = wp0[k + 1];
        b1.x = wp1[k];  b1.y = wp1[k + 1];
        acc0 = __builtin_amdgcn_wmma_f32_16x16x4_f32(false, a, false, b0,
                                                     (short)0, acc0, false, false);
        acc1 = __builtin_amdgcn_wmma_f32_16x16x4_f32(false, a, false, b1,
                                                     (short)0, acc1, false, false);
    }

    const int row0 = m0 + ((lane >> 4) << 3);   // m0 or m0+8
    const int c0   = n0 + r;
    const int c1   = c0 + 16;
    const float bv0 = bias ? bias[c0] : 0.f;
    const float bv1 = bias ? bias[c1] : 0.f;
#pragma unroll
    for (int v = 0; v < 8; ++v) {
        size_t i0 = (size_t)(row0 + v) * ldc + c0;
        size_t i1 = (size_t)(row0 + v) * ldc + c1;
        float v0 = acc0[v] + bv0;
        float v1 = acc1[v] + bv1;
        if (accumulate) { v0 += C[i0]; v1 += C[i1]; }
        C[i0] = v0;
        C[i1] = v1;
    }
}

// =====================================================================
// LDS-staged fp32 WMMA GEMM for the large throughput GEMMs.
// Workgroup = 256 threads (8 waves) computes a 128x128 tile; K step 32.
// A(128x32) and W(128x32) K-slabs are staged into LDS with CDNA5 async
// global->LDS copies (GLOBAL_LOAD_ASYNC_TO_LDS_B128, ASYNCcnt), then each
// wave computes 16 rows x 128 cols via 8 WMMA accumulators fed by ds loads.
// LDS rows padded to 36 floats to break stride-32 bank aliasing.
// Requires M%128==0, N%128==0, K%32==0 (true for all large GEMMs here).
// =====================================================================
#define KB      32
#define LDS_PAD 36

__device__ __forceinline__ void async_ld_b128(unsigned lds_off, const float* gaddr)
{
    asm volatile("global_load_async_to_lds_b128 %0, %1, off"
                 :: "v"(lds_off), "v"(gaddr) : "memory");
}

__global__ __launch_bounds__(256)
void gemm_wmma_f32_tiled(const float* __restrict__ A, int lda,
                         const float* __restrict__ W, int ldw,
                         float* __restrict__ C, int ldc,
                         int K, const float* __restrict__ bias, int accumulate)
{
    __shared__ float sA[128 * LDS_PAD];
    __shared__ float sW[128 * LDS_PAD];

    const int m0   = blockIdx.y << 7;
    const int n0   = blockIdx.x << 7;
    const int t    = threadIdx.x;
    const int lane = t & 31;
    const int wv   = t >> 5;          // wave id 0..7 -> 16-row slice of M tile
    const int r    = lane & 15;
    const int half = lane >> 4;       // 0/1

    v8f acc[8] = {};                  // 8 N-tiles of 16x16

    const unsigned aBase = (unsigned)(size_t)(&sA[0]);
    const unsigned wBase = (unsigned)(size_t)(&sW[0]);

    for (int kb = 0; kb < K; kb += KB) {
        // ---- stage A/W K-slabs into LDS via async copies ----
        // 128 rows x 32 floats = 1024 x b128 segments per matrix;
        // 256 threads issue 4 segments each per matrix.
#pragma unroll
        for (int i = 0; i < 4; ++i) {
            int li  = t * 4 + i;          // 0..1023
            int row = li >> 3;            // 0..127
            int seg = (li & 7) << 2;      // float offset 0,4,..,28
            unsigned lofs = (unsigned)((row * LDS_PAD + seg) << 2);
            const float* ga = A + (size_t)(m0 + row) * lda + kb + seg;
            const float* gw = W + (size_t)(n0 + row) * ldw + kb + seg;
            async_ld_b128(aBase + lofs, ga);
            async_ld_b128(wBase + lofs, gw);
        }
        asm volatile("s_wait_asynccnt 0" ::: "memory");
        __syncthreads();

        // ---- compute on the staged slab ----
#pragma unroll
        for (int kk = 0; kk < KB; kk += 4) {
            const int ac = kk + (half << 1);
            v2f a;
            a.x = sA[(wv * 16 + r) * LDS_PAD + ac];
            a.y = sA[(wv * 16 + r) * LDS_PAD + ac + 1];
#pragma unroll
            for (int j = 0; j < 8; ++j) {
                v2f b;
                b.x = sW[(j * 16 + r) * LDS_PAD + ac];
                b.y = sW[(j * 16 + r) * LDS_PAD + ac + 1];
                acc[j] = __builtin_amdgcn_wmma_f32_16x16x4_f32(false, a, false, b,
                                                               (short)0, acc[j],
                                                               false, false);
            }
        }
        __syncthreads();
    }

    // ---- store 16x128 per wave ----
    const int row0 = m0 + wv * 16 + (half << 3);
#pragma unroll
    for (int j = 0; j < 8; ++j) {
        const int col = n0 + j * 16 + r;
        const float bv = bias ? bias[col] : 0.f;
#pragma unroll
        for (int v = 0; v < 8; ++v) {
            size_t idx = (size_t)(row0 + v) * ldc + col;
            float val = acc[j][v] + bv;
            if (accumulate) val += C[idx];
            C[idx] = val;
        }
    }
}

static inline void launch_gemm(const float* A, int lda, const float* W, int ldw,
                               float* C, int ldc, int M, int N, int K,
                               const float* bias, int accumulate, hipStream_t s)
{
    if ((M % 128 == 0) && (N % 128 == 0) && (K % KB == 0)) {
        dim3 grid(N / 128, M / 128);
        gemm_wmma_f32_tiled<<<grid, 256, 0, s>>>(A, lda, W, ldw, C, ldc, K, bias, accumulate);
    } else {
        dim3 grid(N / 32, M / 16);
        gemm_wmma_f32<<<grid, 32, 0, s>>>(A, lda, W, ldw, C, ldc, K, bias, accumulate);
    }
}

// ---------------- embedding gather ----------------
__global__ void gather_rows(const int* __restrict__ idx, const float* __restrict__ table,
                            float* __restrict__ out, int ntok, int E)
{
    size_t i = (size_t)blockIdx.x * blockDim.x + threadIdx.x;
    if (i >= (size_t)ntok * E) return;
    int t = (int)(i / E);
    int e = (int)(i % E);
    out[i] = table[(size_t)idx[t] * E + e];
}

// ---------------- GRU gates ----------------
__global__ void gru_gate(const float* __restrict__ gi_a, const float* __restrict__ gi_b,
                         const float* __restrict__ gh,
                         const float* __restrict__ bi, const float* __restrict__ bh,
                         const float* __restrict__ h_in, int h_in_stride,
                         float* __restrict__ h_out, int h_out_stride)
{
    int t = blockIdx.x * blockDim.x + threadIdx.x;
    if (t >= B_DIM * H_DIM) return;
    int b = t / H_DIM, j = t % H_DIM;
    size_t g0 = (size_t)b * 3 * H_DIM + j;

    float ir = gi_a[g0]            + bi[j];
    float iz = gi_a[g0 + H_DIM]    + bi[j + H_DIM];
    float in = gi_a[g0 + 2*H_DIM]  + bi[j + 2*H_DIM];
    if (gi_b) {
        ir += gi_b[g0];
        iz += gi_b[g0 + H_DIM];
        in += gi_b[g0 + 2*H_DIM];
    }
    float hr = bh[j], hz = bh[j + H_DIM], hn = bh[j + 2*H_DIM];
    if (gh) {
        hr += gh[g0];
        hz += gh[g0 + H_DIM];
        hn += gh[g0 + 2*H_DIM];
    }
    float r = 1.f / (1.f + __expf(-(ir + hr)));
    float z = 1.f / (1.f + __expf(-(iz + hz)));
    float n = tanhf(in + r * hn);
    float hp = h_in ? h_in[(size_t)b * h_in_stride + j] : 0.f;
    h_out[(size_t)b * h_out_stride + j] = (1.f - z) * n + z * hp;
}

// ---------------- attention ----------------
__global__ void attn_scores(const float* __restrict__ q, const float* __restrict__ keys,
                            const float* __restrict__ v, const float* __restrict__ mask,
                            float* __restrict__ scores)
{
    int wave = (blockIdx.x * blockDim.x + threadIdx.x) >> 5;
    int lane = threadIdx.x & 31;
    if (wave >= S_DIM * B_DIM) return;
    int s = wave / B_DIM, b = wave % B_DIM;
    const float* kp = keys + ((size_t)s * B_DIM + b) * H_DIM;
    const float* qp = q + (size_t)b * H_DIM;
    float acc = 0.f;
    for (int j = lane; j < H_DIM; j += 32)
        acc += tanhf(qp[j] + kp[j]) * v[j];
    for (int off = 16; off > 0; off >>= 1)
        acc += __shfl_down(acc, off);
    if (lane == 0) {
        float m = mask[s * B_DIM + b];
        scores[s * B_DIM + b] = (m > 0.f) ? acc : -1.0e9f;
    }
}

__global__ void attn_softmax(float* __restrict__ scores)
{
    int b = threadIdx.x;
    if (b >= B_DIM) return;
    float mx = -1.0e30f;
    for (int s = 0; s < S_DIM; ++s) mx = fmaxf(mx, scores[s * B_DIM + b]);
    float sum = 0.f;
    for (int s = 0; s < S_DIM; ++s) {
        float e = __expf(scores[s * B_DIM + b] - mx);
        scores[s * B_DIM + b] = e;
        sum += e;
    }
    float inv = 1.f / sum;
    for (int s = 0; s < S_DIM; ++s) scores[s * B_DIM + b] *= inv;
}

__global__ void attn_ctx(const float* __restrict__ alpha, const float* __restrict__ enc,
                         float* __restrict__ ctx)
{
    int t = blockIdx.x * blockDim.x + threadIdx.x;
    const int E2 = 2 * H_DIM;
    if (t >= B_DIM * E2) return;
    int b = t / E2, e = t % E2;
    float acc = 0.f;
    for (int s = 0; s < S_DIM; ++s)
        acc += alpha[s * B_DIM + b] * enc[((size_t)s * B_DIM + b) * E2 + e];
    ctx[(size_t)b * E2 + e] = acc;
}

// ---------------- readout prep ----------------
__global__ void build_final_inputs(const float* __restrict__ dec_states,
                                   const float* __restrict__ te,
                                   const float* __restrict__ fwd_ctx,
                                   const float* __restrict__ rev_ctx,
                                   float* __restrict__ dstate,
                                   float* __restrict__ near,
                                   float* __restrict__ sctx)
{
    const int Tm2 = T_DIM - 2, E2 = 2 * H_DIM;
    size_t i = (size_t)blockIdx.x * blockDim.x + threadIdx.x;
    if (i >= (size_t)Tm2 * B_DIM * E2) return;
    int j = (int)(i % E2);
    size_t tb = i / E2;
    int b = (int)(tb % B_DIM);
    int t = (int)(tb / B_DIM);

    int trow = (j < H_DIM) ? t : (t + 2);
    dstate[i] = dec_states[((size_t)trow * B_DIM + b) * E2 + j];
    near[i] = (j < ET_DIM)
        ? te[((size_t)t * B_DIM + b) * ET_DIM + j]
        : te[((size_t)(t + 2) * B_DIM + b) * ET_DIM + (j - ET_DIM)];
    size_t ci = ((size_t)(t + 1) * B_DIM + b) * E2 + j;
    sctx[i] = fwd_ctx[ci] + rev_ctx[ci];
}

__global__ void maxout_kernel(const float* __restrict__ act, float* __restrict__ mo, size_t n)
{
    size_t i = (size_t)blockIdx.x * blockDim.x + threadIdx.x;
    if (i >= n) return;
    mo[i] = fmaxf(act[2 * i], act[2 * i + 1]);
}

__global__ void preqv_kernel(const int* __restrict__ target, const float* __restrict__ Wv,
                             const float* __restrict__ oe, float* __restrict__ out)
{
    const int Tm2 = T_DIM - 2;
    size_t i = (size_t)blockIdx.x * blockDim.x + threadIdx.x;
    if (i >= (size_t)Tm2 * B_DIM * EO_DIM) return;
    int e = (int)(i % EO_DIM);
    size_t tb = i / EO_DIM;
    int b = (int)(tb % B_DIM);
    int t = (int)(tb / B_DIM);
    int tok = target[(t + 1) * B_DIM + b];
    out[i] = Wv[(size_t)tok * EO_DIM + e] * oe[i];
}

__global__ void copy_f32(const float* __restrict__ src, float* __restrict__ dst, size_t n)
{
    size_t i = (size_t)blockIdx.x * blockDim.x + threadIdx.x;
    if (i < n) dst[i] = src[i];
}

// =====================================================================
extern "C" void kernel_launch(void* const* d_in, const int* in_sizes, int n_in,
                              void* d_out, int out_size, void* d_ws, size_t ws_size,
                              hipStream_t stream)
{
    const int*   source      = (const int*)  d_in[0];
    const float* source_mask = (const float*)d_in[1];
    const int*   target      = (const int*)  d_in[2];
    const float* src_emb     = (const float*)d_in[3];
    const float* tgt_emb     = (const float*)d_in[4];
    const float* enc_Wi_f    = (const float*)d_in[5];
    const float* enc_Wh_f    = (const float*)d_in[6];
    const float* enc_bi_f    = (const float*)d_in[7];
    const float* enc_bh_f    = (const float*)d_in[8];
    const float* enc_Wi_r    = (const float*)d_in[9];
    const float* enc_Wh_r    = (const float*)d_in[10];
    const float* enc_bi_r    = (const float*)d_in[11];
    const float* enc_bh_r    = (const float*)d_in[12];
    const float* fattn_Wq    = (const float*)d_in[13];
    const float* fattn_bq    = (const float*)d_in[14];
    const float* fattn_Wk    = (const float*)d_in[15];
    const float* fattn_bk    = (const float*)d_in[16];
    const float* fattn_v     = (const float*)d_in[17];
    const float* rattn_Wq    = (const float*)d_in[18];
    const float* rattn_bq    = (const float*)d_in[19];
    const float* rattn_Wk    = (const float*)d_in[20];
    const float* rattn_bk    = (const float*)d_in[21];
    const float* rattn_v     = (const float*)d_in[22];
    const float* dec_Wi_f    = (const float*)d_in[23];
    const float* dec_Wh_f    = (const float*)d_in[24];
    const float* dec_bi_f    = (const float*)d_in[25];
    const float* dec_bh_f    = (const float*)d_in[26];
    const float* dec_Wi_r    = (const float*)d_in[27];
    const float* dec_Wh_r    = (const float*)d_in[28];
    const float* dec_bi_r    = (const float*)d_in[29];
    const float* dec_bh_r    = (const float*)d_in[30];
    const float* Wd          = (const float*)d_in[31];
    const float* bd          = (const float*)d_in[32];
    const float* Ww          = (const float*)d_in[33];
    const float* bw          = (const float*)d_in[34];
    const float* Wc          = (const float*)d_in[35];
    const float* bc          = (const float*)d_in[36];
    const float* Wf          = (const float*)d_in[37];
    const float* Wv          = (const float*)d_in[38];

    const int H2 = 2 * H_DIM, H3 = 3 * H_DIM;
    const int Tm2 = T_DIM - 2;
    const int SB = S_DIM * B_DIM, TB = T_DIM * B_DIM;
    const int dec_in = ET_DIM + H2;   // 1536

    // ---- workspace carve ----
    float* w = (float*)d_ws;
    auto alloc = [&](size_t n) { float* r = w; w += n; return r; };
    float* se       = alloc((size_t)SB * ES_DIM);
    float* gi_enc_f = alloc((size_t)SB * H3);
    float* gi_enc_r = alloc((size_t)SB * H3);
    float* enc      = alloc((size_t)SB * H2);
    float* keys_f   = alloc((size_t)SB * H_DIM);
    float* keys_r   = alloc((size_t)SB * H_DIM);
    float* te       = alloc((size_t)TB * ET_DIM);
    float* te_gi_f  = alloc((size_t)TB * H3);
    float* te_gi_r  = alloc((size_t)TB * H3);
    float* dec_st   = alloc((size_t)TB * H2);
    float* fwd_ctx  = alloc((size_t)TB * H2);
    float* rev_ctx  = alloc((size_t)TB * H2);
    float* gh_buf   = alloc((size_t)B_DIM * H3);
    float* gic_buf  = alloc((size_t)B_DIM * H3);
    float* q_buf    = alloc((size_t)B_DIM * H_DIM);
    float* scores   = alloc((size_t)SB);
    float* dstate   = alloc((size_t)Tm2 * B_DIM * H2);
    float* near     = alloc((size_t)Tm2 * B_DIM * H2);
    float* sctx     = alloc((size_t)Tm2 * B_DIM * H2);
    float* act      = alloc((size_t)Tm2 * B_DIM * 2 * M_DIM);
    float* mo       = alloc((size_t)Tm2 * B_DIM * M_DIM);
    float* oe       = alloc((size_t)Tm2 * B_DIM * EO_DIM);

    auto ew_grid = [](size_t n) { return dim3((unsigned)((n + 255) / 256)); };

    // ---- 1. embeddings ----
    gather_rows<<<ew_grid((size_t)SB * ES_DIM), 256, 0, stream>>>(source, src_emb, se, SB, ES_DIM);
    gather_rows<<<ew_grid((size_t)TB * ET_DIM), 256, 0, stream>>>(target, tgt_emb, te, TB, ET_DIM);

    // ---- 2. encoder input gates precompute (biases added in gate) ----
    launch_gemm(se, ES_DIM, enc_Wi_f, ES_DIM, gi_enc_f, H3, SB, H3, ES_DIM, nullptr, 0, stream);
    launch_gemm(se, ES_DIM, enc_Wi_r, ES_DIM, gi_enc_r, H3, SB, H3, ES_DIM, nullptr, 0, stream);

    const dim3 gate_grid(((B_DIM * H_DIM) + 255) / 256), gate_blk(256);

    // ---- 3. encoder forward (writes enc[:, :, 0:H]) ----
    for (int t = 0; t < S_DIM; ++t) {
        const float* h_prev = (t > 0) ? enc + (size_t)(t - 1) * B_DIM * H2 : nullptr;
        if (t > 0)
            launch_gemm(h_prev, H2, enc_Wh_f, H_DIM, gh_buf, H3, B_DIM, H3, H_DIM, nullptr, 0, stream);
        gru_gate<<<gate_grid, gate_blk, 0, stream>>>(
            gi_enc_f + (size_t)t * B_DIM * H3, nullptr, (t > 0) ? gh_buf : nullptr,
            enc_bi_f, enc_bh_f, h_prev, H2,
            enc + (size_t)t * B_DIM * H2, H2);
    }
    // ---- 4. encoder reverse (writes enc[:, :, H:2H]) ----
    for (int t = S_DIM - 1; t >= 0; --t) {
        bool first = (t == S_DIM - 1);
        const float* h_prev = first ? nullptr : enc + (size_t)(t + 1) * B_DIM * H2 + H_DIM;
        if (!first)
            launch_gemm(h_prev, H2, enc_Wh_r, H_DIM, gh_buf, H3, B_DIM, H3, H_DIM, nullptr, 0, stream);
        gru_gate<<<gate_grid, gate_blk, 0, stream>>>(
            gi_enc_r + (size_t)t * B_DIM * H3, nullptr, first ? nullptr : gh_buf,
            enc_bi_r, enc_bh_r, h_prev, H2,
            enc + (size_t)t * B_DIM * H2 + H_DIM, H2);
    }

    // ---- 5. attention keys ----
    launch_gemm(enc, H2, fattn_Wk, H2, keys_f, H_DIM, SB, H_DIM, H2, fattn_bk, 0, stream);
    launch_gemm(enc, H2, rattn_Wk, H2, keys_r, H_DIM, SB, H_DIM, H2, rattn_bk, 0, stream);

    // ---- 6. decoder input gates precompute (embedding columns of dec_Wi) ----
    launch_gemm(te, ET_DIM, dec_Wi_f, dec_in, te_gi_f, H3, TB, H3, ET_DIM, nullptr, 0, stream);
    launch_gemm(te, ET_DIM, dec_Wi_r, dec_in, te_gi_r, H3, TB, H3, ET_DIM, nullptr, 0, stream);

    const dim3 sc_grid((S_DIM * B_DIM * 32 + 255) / 256), sc_blk(256);
    const dim3 cx_grid((B_DIM * H2 + 255) / 256), cx_blk(256);

    // ---- 7. decoder forward ----
    gru_gate<<<gate_grid, gate_blk, 0, stream>>>(
        te_gi_f, nullptr, nullptr, dec_bi_f, dec_bh_f, nullptr, H2, dec_st, H2);
    for (int t = 1; t < T_DIM; ++t) {
        const float* h_prev = dec_st + (size_t)(t - 1) * B_DIM * H2;
        float* ctx = fwd_ctx + (size_t)t * B_DIM * H2;
        launch_gemm(h_prev, H2, fattn_Wq, H_DIM, q_buf, H_DIM, B_DIM, H_DIM, H_DIM, fattn_bq, 0, stream);
        attn_scores<<<sc_grid, sc_blk, 0, stream>>>(q_buf, keys_f, fattn_v, source_mask, scores);
        attn_softmax<<<1, 64, 0, stream>>>(scores);
        attn_ctx<<<cx_grid, cx_blk, 0, stream>>>(scores, enc, ctx);
        launch_gemm(ctx, H2, dec_Wi_f + ET_DIM, dec_in, gic_buf, H3, B_DIM, H3, H2, nullptr, 0, stream);
        launch_gemm(h_prev, H2, dec_Wh_f, H_DIM, gh_buf, H3, B_DIM, H3, H_DIM, nullptr, 0, stream);
        gru_gate<<<gate_grid, gate_blk, 0, stream>>>(
            te_gi_f + (size_t)t * B_DIM * H3, gic_buf, gh_buf,
            dec_bi_f, dec_bh_f, h_prev, H2,
            dec_st + (size_t)t * B_DIM * H2, H2);
    }

    // ---- 8. decoder reverse ----
    gru_gate<<<gate_grid, gate_blk, 0, stream>>>(
        te_gi_r + (size_t)(T_DIM - 1) * B_DIM * H3, nullptr, nullptr,
        dec_bi_r, dec_bh_r, nullptr, H2,
        dec_st + (size_t)(T_DIM - 1) * B_DIM * H2 + H_DIM, H2);
    for (int t = T_DIM - 2; t >= 0; --t) {
        const float* h_prev = dec_st + (size_t)(t + 1) * B_DIM * H2 + H_DIM;
        float* ctx = rev_ctx + (size_t)t * B_DIM * H2;
        launch_gemm(h_prev, H2, rattn_Wq, H_DIM, q_buf, H_DIM, B_DIM, H_DIM, H_DIM, rattn_bq, 0, stream);
        attn_scores<<<sc_grid, sc_blk, 0, stream>>>(q_buf, keys_r, rattn_v, source_mask, scores);
        attn_softmax<<<1, 64, 0, stream>>>(scores);
        attn_ctx<<<cx_grid, cx_blk, 0, stream>>>(scores, enc, ctx);
        launch_gemm(ctx, H2, dec_Wi_r + ET_DIM, dec_in, gic_buf, H3, B_DIM, H3, H2, nullptr, 0, stream);
        launch_gemm(h_prev, H2, dec_Wh_r, H_DIM, gh_buf, H3, B_DIM, H3, H_DIM, nullptr, 0, stream);
        gru_gate<<<gate_grid, gate_blk, 0, stream>>>(
            te_gi_r + (size_t)t * B_DIM * H3, gic_buf, gh_buf,
            dec_bi_r, dec_bh_r, h_prev, H2,
            dec_st + (size_t)t * B_DIM * H2 + H_DIM, H2);
    }

    // ---- 9. readout ----
    const size_t nfin = (size_t)Tm2 * B_DIM * H2;
    build_final_inputs<<<ew_grid(nfin), 256, 0, stream>>>(dec_st, te, fwd_ctx, rev_ctx,
                                                          dstate, near, sctx);
    const int Mrows = Tm2 * B_DIM;               // 3072
    launch_gemm(dstate, H2, Wd, H2,       act, 2 * M_DIM, Mrows, 2 * M_DIM, H2, bd, 0, stream);
    launch_gemm(near,   H2, Ww, 2*ET_DIM, act, 2 * M_DIM, Mrows, 2 * M_DIM, 2*ET_DIM, bw, 1, stream);
    launch_gemm(sctx,   H2, Wc, H2,       act, 2 * M_DIM, Mrows, 2 * M_DIM, H2, bc, 1, stream);

    const size_t nmo = (size_t)Mrows * M_DIM;
    maxout_kernel<<<ew_grid(nmo), 256, 0, stream>>>(act, mo, nmo);
    launch_gemm(mo, M_DIM, Wf, M_DIM, oe, EO_DIM, Mrows, EO_DIM, M_DIM, nullptr, 0, stream);

    // ---- 10. outputs ----
    float* out_final  = (float*)d_out;
    float* out_preqv  = out_final + (size_t)Mrows * VT_DIM;
    float* out_postqv = out_preqv + (size_t)Mrows * EO_DIM;

    launch_gemm(oe, EO_DIM, Wv, EO_DIM, out_final, VT_DIM, Mrows, VT_DIM, EO_DIM, nullptr, 0, stream);

    const size_t npre = (size_t)Mrows * EO_DIM;
    preqv_kernel<<<ew_grid(npre), 256, 0, stream>>>(target, Wv, oe, out_preqv);

    const size_t npost = (size_t)Mrows * H2;
    copy_f32<<<ew_grid(npost), 256, 0, stream>>>(dec_st + (size_t)B_DIM * H2, out_postqv, npost);
}